// fma_26182120636582
// MI455X (gfx1250) — compile-verified
//
#include <hip/hip_runtime.h>
#include <cstdint>
#include <cstddef>

// ---------------------------------------------------------------------------
// FMA (fast-multipole attention) forward for MI455X / gfx1250, fp32 WMMA path.
//   B=32, N=4096, D=64, fine block M=32, P=4 summaries per downsample block,
//   6 downsample levels (w = 32<<l), fine offsets {-1,0,1}, coarse {-3,-2,2,3}.
// All matrix math uses V_WMMA_F32_16X16X4_F32 (wave32, fp32 in/out) to keep
// reference fp32 numerics. Bandwidth-bound problem -> fused attention kernel
// keeps the 32x288 score panel entirely in LDS.
// ---------------------------------------------------------------------------

typedef __attribute__((ext_vector_type(2))) float v2f;
typedef __attribute__((ext_vector_type(8))) float v8f;

static constexpr int kB     = 32;
static constexpr int kN     = 4096;
static constexpr int kD     = 64;
static constexpr int kMF    = 32;            // fine block
static constexpr int kNB0   = kN / kMF;      // 128 fine blocks
static constexpr int kNFINE = 96;            // 3 fine neighbor blocks * 32
static constexpr int kNCOLS = 96 + 6 * 32;   // + 6 levels * 4 blocks * 8 summaries
// kls[l] (l=1..6) packed region: level l holds 32 * (1024>>l) * 64 floats.
// cumulative offset: lvloff(l) = 2097152 - (2097152 >> (l-1))
static constexpr size_t kVlsBase = 2097152;  // vls region starts here (floats)

__device__ __forceinline__ v8f wmma_f32x4(v2f a, v2f b, v8f c) {
  // D = A(16x4) * B(4x16) + C(16x16), all fp32
  return __builtin_amdgcn_wmma_f32_16x16x4_f32(false, a, false, b, (short)0, c,
                                               false, false);
}

// ---------------------------------------------------------------------------
// Downsample (strided conv == GEMM):  C[o, g] = sum_r W[o, r] * X[r, g] + bias
//   o in [0,256): o = di*4 + j ; g = b*nb + t ; r = wi*64 + c  (so that the
//   X column is the contiguous memory chunk src[(b*4096 + t*w)*64 + r]).
// One workgroup per 16x16 C tile; 8 waves split the reduction, partials are
// combined through LDS in a fixed order (deterministic).
// K-loop is strength-reduced: within a 64-wide chunk wi is constant, the W
// pointer steps by 4*w and the src pointer by 4; across chunks W steps by 1.
// ---------------------------------------------------------------------------
__global__ __launch_bounds__(256) void fma_downsample_wmma(
    const float* __restrict__ src,   // (32, 4096, 64)
    const float* __restrict__ W,     // (256, 64, w) row-major
    const float* __restrict__ bias,  // (256)
    float* __restrict__ dst,         // (32, nb*4, 64)
    int w, int nb, int Kred)
{
  __shared__ float part[8][256];
  const int tid  = threadIdx.x;
  const int lane = tid & 31;
  const int wv   = __builtin_amdgcn_readfirstlane(tid >> 5);  // wave-uniform SGPR
  const int half = lane >> 4;
  const int l15  = lane & 15;
  const int gtile = blockIdx.x;   // G/16 tiles over g
  const int otile = blockIdx.y;   // 16 tiles over o

  // per-lane B column (contiguous chunk of src)
  const int g  = gtile * 16 + l15;
  const int bb = g / nb;
  const int tt = g - bb * nb;
  const float* colp = src + ((size_t)bb * kN + (size_t)tt * w) * kD;
  // per-lane A row of W
  const int o = otile * 16 + l15;
  const float* wrow = W + (size_t)o * Kred;

  const int slice = Kred >> 3;       // Kred = 64*w, w>=32 -> multiple of 256
  const int lo = wv * slice;
  const int hi = lo + slice;

  const float* bp  = colp + lo + half * 2;                    // src stream
  const float* ap0 = wrow + (size_t)(half * 2) * w + (lo >> 6); // W chunk base
  const size_t wstep = 4 * (size_t)w;

  v8f acc = {};
  for (int chunk = lo; chunk < hi; chunk += 64) {
    const float* ap = ap0;
    __builtin_prefetch(bp + 256, 0, 1);   // src: 4 chunks ahead
    __builtin_prefetch(ap0 + 8, 0, 1);    // W: next wi cachelines
#pragma unroll
    for (int i = 0; i < 16; ++i) {        // 16 k-steps of 4 per chunk
      v2f bfr; bfr[0] = bp[0]; bfr[1] = bp[1];
      v2f afr; afr[0] = ap[0]; afr[1] = ap[w];
      acc = wmma_f32x4(afr, bfr, acc);
      ap += wstep;                        // c0 += 4, same wi
      bp += 4;
    }
    ap0 += 1;                             // wi += 1
  }
#pragma unroll
  for (int vv = 0; vv < 8; ++vv) part[wv][vv * 32 + lane] = acc[vv];
  __syncthreads();

  // fixed-order cross-wave reduction: one C element per thread
  const int ve = tid >> 5;
  const int ln = tid & 31;
  float s = 0.f;
#pragma unroll
  for (int wvi = 0; wvi < 8; ++wvi) s += part[wvi][ve * 32 + ln];
  const int hh = ln >> 4, ll = ln & 15;
  const int orow = otile * 16 + ve + hh * 8;   // o index of this element
  const int gg   = gtile * 16 + ll;
  const int b2 = gg / nb;
  const int t2 = gg - b2 * nb;
  const int di = orow >> 2, jj = orow & 3;     // o = di*4 + jj
  dst[(((size_t)b2 * nb + t2) * 4 + jj) * kD + di] = s + bias[orow];
}

// ---------------------------------------------------------------------------
// Gather the 288 K (or V) rows for fine block g0 into LDS; OOB blocks -> 0
// (matches the reference's untouched zero score buffer semantics).
// ---------------------------------------------------------------------------
__device__ __forceinline__ void load_kv_panel(
    float* __restrict__ KV, const float* __restrict__ fine,
    const float* __restrict__ coarse, int b, int g0, int tid)
{
  for (int idx = tid; idx < kNCOLS * (kD / 4); idx += 256) {
    const int row = idx >> 4;
    const int c4  = (idx & 15) * 4;
    float4 val; val.x = val.y = val.z = val.w = 0.f;
    if (row < kNFINE) {
      const int srcb = g0 + (row >> 5) - 1;          // offsets -1,0,1
      if (srcb >= 0 && srcb < kNB0) {
        val = *(const float4*)(fine +
              ((size_t)b * kN + srcb * kMF + (row & 31)) * kD + c4);
      }
    } else {
      const int r2     = row - kNFINE;
      const int l      = 1 + (r2 >> 5);              // level 1..6
      const int within = r2 & 31;
      const int sI     = within >> 3;                // which coarse offset
      const int p      = within & 7;                 // summary within block
      const int off    = (sI == 0) ? -3 : (sI == 1) ? -2 : (sI == 2) ? 2 : 3;
      const int gl     = g0 >> l;
      const int nbl    = kNB0 >> l;
      const int srcb   = gl + off;
      if (srcb >= 0 && srcb < nbl) {
        const size_t lvloff = 2097152u - (2097152u >> (l - 1));
        const int R = 1024 >> l;                     // rows per batch at level l
        val = *(const float4*)(coarse + lvloff +
              ((size_t)b * R + srcb * 8 + p) * kD + c4);
      }
    }
    *(float4*)(KV + row * kD + c4) = val;
  }
}

// ---------------------------------------------------------------------------
// Fused stage1 + multi-level softmax + stage2. One workgroup (8 waves) per
// (batch, fine block). Dynamic LDS ~118KB (WGP has 320KB).
// ---------------------------------------------------------------------------
__global__ __launch_bounds__(256) void fma_fused_attention(
    const float* __restrict__ q, const float* __restrict__ k,
    const float* __restrict__ v, const float* __restrict__ kls,
    const float* __restrict__ vls, float* __restrict__ out)
{
  extern __shared__ float smem[];
  float* Qs   = smem;                   // 32*64
  float* KV   = Qs + kMF * kD;          // 288*64 (K panel, later V panel)
  float* S    = KV + kNCOLS * kD;       // 32*288 scores
  float* maxb = S + kMF * kNCOLS;       // 7*32 per-(row,level) max
  float* rmin = maxb + 224;             // 32
  float* rinv = rmin + 32;              // 32
  float* sumb = rinv + 32;              // 7*32

  const int g0   = blockIdx.x;          // fine block 0..127
  const int b    = blockIdx.y;          // batch 0..31
  const int tid  = threadIdx.x;
  const int lane = tid & 31;
  const int wv   = __builtin_amdgcn_readfirstlane(tid >> 5);  // SGPR wave id
  const int half = lane >> 4;
  const int l15  = lane & 15;

  // Q tile, pre-scaled by 1/sqrt(64)
  for (int idx = tid; idx < kMF * (kD / 4); idx += 256) {
    const int row = idx >> 4;
    const int c4  = (idx & 15) * 4;
    float4 qv = *(const float4*)(q + ((size_t)b * kN + g0 * kMF + row) * kD + c4);
    qv.x *= 0.125f; qv.y *= 0.125f; qv.z *= 0.125f; qv.w *= 0.125f;
    *(float4*)(Qs + row * kD + c4) = qv;
  }
  load_kv_panel(KV, k, kls, b, g0, tid);
  __syncthreads();

  // ---- stage 1: S(32x288) = Qs(32x64) @ KV^T, 2x18 WMMA tiles over 8 waves
  for (int t = wv; t < 36; t += 8) {
    const int mr = (t / 18) * 16;
    const int nc = (t % 18) * 16;
    v8f acc = {};
#pragma unroll
    for (int kb = 0; kb < kD; kb += 4) {
      const int k0 = kb + half * 2;
      const float* ap = Qs + (mr + l15) * kD + k0;   // A[m][k]
      const float* bp = KV + (nc + l15) * kD + k0;   // B[k][n] = K[n][k]
      v2f a;  a[0]  = ap[0]; a[1]  = ap[1];
      v2f bb; bb[0] = bp[0]; bb[1] = bp[1];
      acc = wmma_f32x4(a, bb, acc);
    }
#pragma unroll
    for (int vv = 0; vv < 8; ++vv)
      S[(mr + vv + half * 8) * kNCOLS + nc + l15] = acc[vv];
  }
  __syncthreads();

  // ---- multi-level softmax: shift by min over levels of per-level max,
  //      scale level l by (32*2^(l-1))/4, normalize by total sum.
  if (tid < 224) {                       // (row, level) pairs: 32 x 7
    const int r   = tid & 31;
    const int lev = tid >> 5;
    const int c0  = (lev == 0) ? 0 : kNFINE + 32 * (lev - 1);
    const int cn  = (lev == 0) ? kNFINE : 32;
    const float* Sr = S + r * kNCOLS + c0;
    float m = -3.402823466e38f;
    for (int c = 0; c < cn; ++c) m = fmaxf(m, Sr[c]);
    maxb[lev * 32 + r] = m;
  }
  __syncthreads();
  if (tid < 32) {
    float m = maxb[tid];
#pragma unroll
    for (int lev = 1; lev < 7; ++lev) m = fminf(m, maxb[lev * 32 + tid]);
    rmin[tid] = m;
  }
  __syncthreads();
  if (tid < 224) {
    const int r   = tid & 31;
    const int lev = tid >> 5;
    const int c0  = (lev == 0) ? 0 : kNFINE + 32 * (lev - 1);
    const int cn  = (lev == 0) ? kNFINE : 32;
    const float scale = (lev == 0) ? 1.0f : (float)(8 << (lev - 1));
    const float mm = rmin[r];
    float* Sr = S + r * kNCOLS + c0;
    float acc = 0.f;
    for (int c = 0; c < cn; ++c) {
      const float e = scale * expf(Sr[c] - mm);
      Sr[c] = e;
      acc += e;
    }
    sumb[lev * 32 + r] = acc;
  }
  __syncthreads();
  if (tid < 32) {
    float s = 0.f;
#pragma unroll
    for (int lev = 0; lev < 7; ++lev) s += sumb[lev * 32 + tid];
    rinv[tid] = 1.0f / s;
  }
  __syncthreads();
  for (int idx = tid; idx < kMF * kNCOLS; idx += 256)
    S[idx] *= rinv[idx / kNCOLS];

  // ---- swap K panel for V panel (everyone is past stage1 reads of KV)
  load_kv_panel(KV, v, vls, b, g0, tid);
  __syncthreads();

  // ---- stage 2: Out(32x64) = S(32x288) @ V(288x64), 2x4 tiles, 1 per wave
  {
    const int mr = (wv >> 2) * 16;
    const int nc = (wv & 3) * 16;
    v8f acc = {};
    for (int kb = 0; kb < kNCOLS; kb += 4) {
      const int k0 = kb + half * 2;
      const float* ap = S + (mr + l15) * kNCOLS + k0;
      v2f a;  a[0]  = ap[0]; a[1]  = ap[1];
      v2f bb;
      bb[0] = KV[(k0 + 0) * kD + nc + l15];
      bb[1] = KV[(k0 + 1) * kD + nc + l15];
      acc = wmma_f32x4(a, bb, acc);
    }
#pragma unroll
    for (int vv = 0; vv < 8; ++vv) {
      const int row = g0 * kMF + mr + vv + half * 8;
      out[((size_t)b * kN + row) * kD + nc + l15] = acc[vv];
    }
  }
}

// ---------------------------------------------------------------------------
// Host launcher.
// Inputs (dict order, lists flattened): 0:q 1:k 2:v 3..8:Wk[0..5] 9..14:bk
// 15..20:Wv[0..5] 21..26:bv. Fallback path if lists arrive concatenated.
// Workspace: kls levels at ws[0..], vls levels at ws[kVlsBase..] (~16MB).
// ---------------------------------------------------------------------------
extern "C" void kernel_launch(void* const* d_in, const int* in_sizes, int n_in,
                              void* d_out, int out_size, void* d_ws, size_t ws_size,
                              hipStream_t stream)
{
  const float* q = (const float*)d_in[0];
  const float* k = (const float*)d_in[1];
  const float* v = (const float*)d_in[2];
  float* out = (float*)d_out;
  float* ws  = (float*)d_ws;

  const float *Wk[6], *bk[6], *Wv[6], *bv[6];
  if (n_in >= 27) {
    for (int l = 0; l < 6; ++l) {
      Wk[l] = (const float*)d_in[3 + l];
      bk[l] = (const float*)d_in[9 + l];
      Wv[l] = (const float*)d_in[15 + l];
      bv[l] = (const float*)d_in[21 + l];
    }
  } else {  // lists concatenated into single buffers
    const float* Wkc = (const float*)d_in[3];
    const float* bkc = (const float*)d_in[4];
    const float* Wvc = (const float*)d_in[5];
    const float* bvc = (const float*)d_in[6];
    size_t woff = 0;
    for (int l = 0; l < 6; ++l) {
      Wk[l] = Wkc + woff; Wv[l] = Wvc + woff;
      bk[l] = bkc + 256 * l; bv[l] = bvc + 256 * l;
      woff += (size_t)256 * 64 * (32u << l);
    }
  }

  // 12 downsample GEMMs (K and V at each level)
  for (int l = 0; l < 6; ++l) {
    const int w = 32 << l, nb = 4096 / w, Kred = 64 * w, G = 32 * nb;
    const size_t off = 2097152u - (2097152u >> l);   // kls[l+1] offset (floats)
    dim3 grid(G / 16, 16);
    fma_downsample_wmma<<<grid, 256, 0, stream>>>(k, Wk[l], bk[l], ws + off,
                                                  w, nb, Kred);
    fma_downsample_wmma<<<grid, 256, 0, stream>>>(v, Wv[l], bv[l],
                                                  ws + kVlsBase + off,
                                                  w, nb, Kred);
  }

  // fused attention: one WG per (fine block, batch)
  const size_t smem_bytes =
      (size_t)(kMF * kD + kNCOLS * kD + kMF * kNCOLS + 224 + 32 + 32 + 224) *
      sizeof(float);
  (void)hipFuncSetAttribute((const void*)fma_fused_attention,
                            hipFuncAttributeMaxDynamicSharedMemorySize,
                            (int)smem_bytes);
  fma_fused_attention<<<dim3(kNB0, kB), 256, smem_bytes, stream>>>(
      q, k, v, ws, ws + kVlsBase, out);
}